// Router_14869176779097
// MI455X (gfx1250) — compile-verified
//
#include <hip/hip_runtime.h>

typedef __attribute__((ext_vector_type(2))) float v2f;
typedef __attribute__((ext_vector_type(8))) float v8f;

#define DDIM   4096
#define NEXP   8
#define NWAVES 4
#define KCHUNK (DDIM / NWAVES)   // 1024 k-values per wave
#define UNROLL 16                // 16 WMMAs (K=64) per unrolled body

__global__ __launch_bounds__(32 * NWAVES)
void router_wmma_kernel(const float* __restrict__ inputs,
                        const float* __restrict__ gate_w,
                        float* __restrict__ out_combine,
                        int* __restrict__ out_idx)
{
    __shared__ float red[NWAVES][32][8];   // partial C tiles, 4 KB

    const int tid  = threadIdx.x;
    const int wave = tid >> 5;
    const int lane = tid & 31;
    const int row  = lane & 15;   // token-in-tile (A) / expert column (B)
    const int hi   = lane >> 4;   // 0 -> k,k+1 ; 1 -> k+2,k+3
    const long tokenBase = (long)blockIdx.x * 16;

    // A: 16x4 f32 tile, per-lane contiguous float2 (non-temporal: streamed once).
    const float* aPtr = inputs + (tokenBase + row) * (long)DDIM + wave * KCHUNK + 2 * hi;
    // B: 4x16 f32 tile from gate_w. Experts 8..15 don't exist; those lanes just
    // reload expert (row & 7), so C columns 8..15 are duplicates of 0..7 that the
    // epilogue never reads. Fully convergent -> no exec predication, no cndmask.
    const float* bPtr = gate_w + (long)(row & (NEXP - 1)) * DDIM + wave * KCHUNK + 2 * hi;

    v8f c = {};
    for (int kk = 0; kk < KCHUNK; kk += 4 * UNROLL) {
        v2f a[UNROLL], b[UNROLL];
#pragma unroll
        for (int u = 0; u < UNROLL; ++u) {
            a[u] = __builtin_nontemporal_load((const v2f*)(aPtr + kk + 4 * u));
        }
#pragma unroll
        for (int u = 0; u < UNROLL; ++u) {
            b[u] = *(const v2f*)(bPtr + kk + 4 * u);
        }
#pragma unroll
        for (int u = 0; u < UNROLL; ++u) {
            // D = A(16x4) x B(4x16) + C, fp32 WMMA
            c = __builtin_amdgcn_wmma_f32_16x16x4_f32(
                    /*neg_a=*/false, a[u], /*neg_b=*/false, b[u],
                    /*c_mod=*/(short)0, c, /*reuse_a=*/false, /*reuse_b=*/false);
        }
    }

#pragma unroll
    for (int r = 0; r < 8; ++r) red[wave][lane][r] = c[r];
    __syncthreads();

    // Epilogue: one thread per token (16 tokens per block)
    if (tid < 16) {
        const int m = tid;
        const int laneSel = (m >> 3) ? 16 : 0;  // logit(m,e): lane e+16*(m>=8), vgpr m&7
        const int r = m & 7;

        float l[NEXP];
#pragma unroll
        for (int e = 0; e < NEXP; ++e) {
            l[e] = red[0][e + laneSel][r] + red[1][e + laneSel][r]
                 + red[2][e + laneSel][r] + red[3][e + laneSel][r];
        }

        float mx = l[0];
#pragma unroll
        for (int e = 1; e < NEXP; ++e) mx = fmaxf(mx, l[e]);
        float p[NEXP];
#pragma unroll
        for (int e = 0; e < NEXP; ++e) p[e] = expf(l[e] - mx);

        // top-2 (strict > keeps the lower index on ties, like jax.lax.top_k)
        int i1 = 0; float p1 = p[0];
#pragma unroll
        for (int e = 1; e < NEXP; ++e) if (p[e] > p1) { p1 = p[e]; i1 = e; }
        int i2 = -1; float p2 = -1.0f;
#pragma unroll
        for (int e = 0; e < NEXP; ++e) if (e != i1 && p[e] > p2) { p2 = p[e]; i2 = e; }

        const float inv = 1.0f / (p1 + p2);
        const long n = tokenBase + m;
#pragma unroll
        for (int e = 0; e < NEXP; ++e) {
            float w = (e == i1) ? p1 * inv : ((e == i2) ? p2 * inv : 0.0f);
            out_combine[n * NEXP + e] = w;
        }
        out_idx[n * 2 + 0] = i1;
        out_idx[n * 2 + 1] = i2;
    }
}

extern "C" void kernel_launch(void* const* d_in, const int* in_sizes, int n_in,
                              void* d_out, int out_size, void* d_ws, size_t ws_size,
                              hipStream_t stream) {
    const float* inputs = (const float*)d_in[0];
    const float* gate_w = (const float*)d_in[1];

    const int D = in_sizes[1] / NEXP;      // 4096
    const int N = in_sizes[0] / D;         // 8192
    (void)n_in; (void)out_size; (void)d_ws; (void)ws_size;

    float* out_combine = (float*)d_out;
    int*   out_idx     = (int*)((float*)d_out + (size_t)N * NEXP);

    dim3 block(32 * NWAVES);
    dim3 grid(N / 16);
    router_wmma_kernel<<<grid, block, 0, stream>>>(inputs, gate_w, out_combine, out_idx);
}